// MultiLayerJKNet_91216515432581
// MI455X (gfx1250) — compile-verified
//
#include <hip/hip_runtime.h>

// ---------------------------------------------------------------------------
// MultiLayerJKNet (GCN x6 + JK-concat + linear head) for MI455X / gfx1250.
//  * fp32 WMMA (V_WMMA_F32_16X16X4_F32) GEMM; B is staged TRANSPOSED through
//    LDS by the TDM (tensor_load_to_lds) with hardware LDS padding so each
//    lane's B fragment is one conflict-free ds_load_b64 (no shuffles).
//  * Edge aggregation: CSR-by-destination (built once per call) + pull-style
//    register accumulation fused with norm/bias/BN/ReLU.
// ---------------------------------------------------------------------------

#define NN   10000
#define EE   160000
#define INF  512
#define HH   512
#define LL   6
#define CC   64
#define LH   (LL * HH)   // 3072
#define EPSB 1e-5f

#define MB   80          // GEMM rows per block  (5 waves x 16)
#define NB   64          // GEMM cols per block  (4 WMMA subtiles)
#define KC   32          // K-chunk staged in LDS (double buffered)
#define KCP  34          // padded LDS row stride (TDM pad: 2 dwords / 32) ->
                         // bank-conflict-free ds_load_b64

typedef __attribute__((ext_vector_type(2))) float        v2f;
typedef __attribute__((ext_vector_type(8))) float        v8f;
typedef __attribute__((ext_vector_type(4))) unsigned int u32x4;
typedef __attribute__((ext_vector_type(4))) int          i32x4;
typedef __attribute__((ext_vector_type(8))) int          i32x8;

#if defined(__HIP_DEVICE_COMPILE__) && __has_builtin(__builtin_amdgcn_tensor_load_to_lds)
#define USE_TDM 1
#else
#define USE_TDM 0
#endif

// ---------------------------------------------------------------------------
// TDM: async-load a [NB rows x KC cols] fp32 tile of BT (row stride ldbt
// elements, contiguous along K) into LDS at byte offset lds_off, padding each
// KC-dword row with 2 dwords (pad_interval code 4 = 32 dwords, pad_amount
// code 1 = 2 dwords) -> LDS row stride KCP.  D# per CDNA5 ISA ch.8.
// Must be executed by exactly one wave (TDM ignores EXEC, issues per wave).
// ---------------------------------------------------------------------------
#if USE_TDM
__device__ __forceinline__ void tdm_load_bt_tile(const float* gsrc,
                                                 unsigned lds_off,
                                                 int ldbt, int ncols) {
  unsigned long long ga = (unsigned long long)(uintptr_t)gsrc;
  u32x4 g0;
  g0[0] = 1u;                                     // count=1, user descriptor
  g0[1] = lds_off;                                // lds_addr (bytes)
  g0[2] = (unsigned)(ga & 0xFFFFFFFFu);           // global_addr[31:0]
  g0[3] = ((unsigned)(ga >> 32) & 0x01FFFFFFu)    // global_addr[56:32]
          | 0x80000000u;                          // type=2 ("image")
  i32x8 g1;
  g1[0] = (int)((2u << 16)                        // data_size=2 (4 bytes)
          | (1u << 20)                            // pad_enable
          | (4u << 22)                            // pad_interval: 32 dwords
          | (1u << 25));                          // pad_amount : 2 dwords
  g1[1] = (int)(((unsigned)ldbt & 0xFFFFu) << 16);          // tensor_dim0[15:0]
  g1[2] = (int)((((unsigned)ldbt >> 16) & 0xFFFFu)
          | (((unsigned)ncols & 0xFFFFu) << 16));           // dim0[31:16] | dim1[15:0]
  g1[3] = (int)((((unsigned)ncols >> 16) & 0xFFFFu)
          | ((unsigned)KC << 16));                          // dim1[31:16] | tile_dim0=KC
  g1[4] = (int)(unsigned)NB;                      // tile_dim1=NB, tile_dim2=0
  g1[5] = (int)ldbt;                              // tensor_dim0_stride[31:0]
  g1[6] = 0;
  g1[7] = 0;
  i32x4 z4 = {0, 0, 0, 0};
#if __clang_major__ >= 23
  i32x8 z8 = {0, 0, 0, 0, 0, 0, 0, 0};
  __builtin_amdgcn_tensor_load_to_lds(g0, g1, z4, z4, z8, 0);
#else
  __builtin_amdgcn_tensor_load_to_lds(g0, g1, z4, z4, 0);
#endif
}
#endif

// ---------------------------------------------------------------------------
// Graph preprocessing: degree counts, norms, CSR by destination
// ---------------------------------------------------------------------------
__global__ void count_kernel(const int* __restrict__ src,
                             const int* __restrict__ dst,
                             int* __restrict__ ocnt,
                             int* __restrict__ icnt, int E) {
  int e = blockIdx.x * blockDim.x + threadIdx.x;
  if (e < E) {
    __hip_atomic_fetch_add(&ocnt[src[e]], 1, __ATOMIC_RELAXED, __HIP_MEMORY_SCOPE_AGENT);
    __hip_atomic_fetch_add(&icnt[dst[e]], 1, __ATOMIC_RELAXED, __HIP_MEMORY_SCOPE_AGENT);
  }
}

__global__ void norm_from_cnt(const int* __restrict__ cnt,
                              float* __restrict__ norm, int n) {
  int i = blockIdx.x * blockDim.x + threadIdx.x;
  if (i < n) {
    int c = cnt[i];
    norm[i] = rsqrtf((float)(c < 1 ? 1 : c));
  }
}

// Single-block exclusive scan of icnt -> row_ptr (N+1) and cursor copy.
__global__ void scan_kernel(const int* __restrict__ cnt,
                            int* __restrict__ row_ptr,
                            int* __restrict__ cursor, int n) {
  __shared__ int sh[256];
  __shared__ int carry;
  if (threadIdx.x == 0) carry = 0;
  __syncthreads();
  for (int base = 0; base < n; base += 256) {
    const int i = base + (int)threadIdx.x;
    const int v = (i < n) ? cnt[i] : 0;
    sh[threadIdx.x] = v;
    __syncthreads();
#pragma unroll
    for (int off = 1; off < 256; off <<= 1) {
      const int t = (threadIdx.x >= (unsigned)off) ? sh[threadIdx.x - off] : 0;
      __syncthreads();
      sh[threadIdx.x] += t;
      __syncthreads();
    }
    const int excl = sh[threadIdx.x] - v;
    if (i < n) {
      const int rp = carry + excl;
      row_ptr[i] = rp;
      cursor[i]  = rp;
    }
    __syncthreads();
    if (threadIdx.x == 255) carry += sh[255];
    __syncthreads();
  }
  if (threadIdx.x == 0) row_ptr[n] = carry;   // == E
}

// Counting-sort placement of in-edges grouped by destination.
__global__ void fill_kernel(const int* __restrict__ src,
                            const int* __restrict__ dst,
                            int* __restrict__ cursor,
                            int* __restrict__ col, int E) {
  int e = blockIdx.x * blockDim.x + threadIdx.x;
  if (e < E) {
    int p = __hip_atomic_fetch_add(&cursor[dst[e]], 1, __ATOMIC_RELAXED, __HIP_MEMORY_SCOPE_AGENT);
    col[p] = src[e];
  }
}

// ---------------------------------------------------------------------------
// 32x32 tiled transpose: out[c][r] = in[r][c]. rows/cols multiples of 32.
// ---------------------------------------------------------------------------
__global__ __launch_bounds__(256) void
transpose_kernel(const float* __restrict__ in, int rows, int cols,
                 float* __restrict__ out) {
  __shared__ float tile[32][33];
  const int c0 = blockIdx.x * 32;
  const int r0 = blockIdx.y * 32;
  const int tx = threadIdx.x & 31;
  const int ty = threadIdx.x >> 5;   // 0..7
#pragma unroll
  for (int i = ty; i < 32; i += 8) {
    tile[i][tx] = in[(size_t)(r0 + i) * cols + c0 + tx];
  }
  __syncthreads();
#pragma unroll
  for (int i = ty; i < 32; i += 8) {
    out[(size_t)(c0 + i) * rows + r0 + tx] = tile[tx][i];
  }
}

// ---------------------------------------------------------------------------
// Block-tiled fp32 WMMA GEMM; B supplied TRANSPOSED (BT: [ncols x K] row-major,
// row stride K) and staged through double-buffered LDS by the TDM.
//   C[MB x NB tile] = (rowScale ? diag(s)*A : A) @ B (+bias)
// Block: 160 threads = 5 waves; wave w -> rows m0+16w..+15, 4 N-subtiles.
// All dims divide exactly -> no bounds checks -> EXEC all ones for WMMA.
//
// fp32 WMMA VGPR layouts (ISA 7.12.2):
//   A 16x4 : lane l (l16=l&15, half=l>>4): a = A[m0+l16][k + 2*half + {0,1}]
//   B 4x16 : b = B[k + 2*half + {0,1}][n0+l16]  == BT[n0+l16][k+2*half+{0,1}]
//            -> one aligned ds_load_b64 from the padded LDS tile
//   C 16x16: vgpr r: C[m0 + r + 8*half][n0 + l16]
// ---------------------------------------------------------------------------
__global__ __launch_bounds__(MB / 16 * 32) void
wmma_gemm_f32(const float* __restrict__ A, int lda,
              const float* __restrict__ rowScale,
              const float* __restrict__ BT, int ncols,
              float* __restrict__ C, int ldc,
              const float* __restrict__ bias, int K) {
  __shared__ float Bsm[2][NB][KCP];

  const int lane = threadIdx.x & 31;
  const int wave = threadIdx.x >> 5;
  const int half = lane >> 4;
  const int l16  = lane & 15;

  const int m0 = blockIdx.x * MB + wave * 16;
  const int n0 = blockIdx.y * NB;

  const int   arow = m0 + l16;
  const float s    = rowScale ? rowScale[arow] : 1.0f;
  const float* __restrict__ Ap = A + (size_t)arow * lda + 2 * half;

  v8f acc[4] = {};

  const int  nChunks = K / KC;
  const bool w0      = (wave == 0);

#if USE_TDM
  if (w0) {
    tdm_load_bt_tile(BT + (size_t)n0 * K, (unsigned)(uintptr_t)&Bsm[0][0][0], K, ncols);
  }
#endif

  for (int c = 0; c < nChunks; ++c) {
    const int k0 = c * KC;
#if USE_TDM
    if (w0) __builtin_amdgcn_s_wait_tensorcnt(0);  // chunk c landed in LDS
    __syncthreads();                               // all waves done with buf (c+1)&1
    if (w0 && (c + 1) < nChunks) {
      tdm_load_bt_tile(BT + (size_t)n0 * K + (k0 + KC),
                       (unsigned)(uintptr_t)&Bsm[(c + 1) & 1][0][0], K, ncols);
    }
    const float* __restrict__ Bs = &Bsm[c & 1][0][0];
#else
    __syncthreads();
    for (int i = threadIdx.x; i < NB * KC; i += MB / 16 * 32) {
      const int nr = i / KC, kk = i % KC;
      Bsm[0][nr][kk] = BT[(size_t)(n0 + nr) * K + k0 + kk];
    }
    __syncthreads();
    const float* __restrict__ Bs = &Bsm[0][0][0];
#endif

    __builtin_prefetch(Ap + k0 + KC, 0, 1);        // global_prefetch_b8

#pragma unroll
    for (int kk = 0; kk < KC; kk += 4) {
      v2f a;
      a.x = Ap[k0 + kk]     * s;
      a.y = Ap[k0 + kk + 1] * s;
      const int kb = kk + 2 * half;                // even -> 8B aligned
#pragma unroll
      for (int j = 0; j < 4; ++j) {
        const v2f b = *(const v2f*)(Bs + (j * 16 + l16) * KCP + kb);
        acc[j] = __builtin_amdgcn_wmma_f32_16x16x4_f32(
            false, a, false, b, (short)0, acc[j], false, false);
      }
    }
  }

#pragma unroll
  for (int j = 0; j < 4; ++j) {
    const float bz = bias ? bias[n0 + j * 16 + l16] : 0.0f;
    float* __restrict__ Cp = C + (size_t)(m0 + 8 * half) * ldc + n0 + j * 16 + l16;
#pragma unroll
    for (int r = 0; r < 8; ++r) {
      Cp[(size_t)r * ldc] = acc[j][r] + bz;
    }
  }
}

// ---------------------------------------------------------------------------
// Pull-style aggregation fused with norm/bias/BN/ReLU.
// One block (256 threads) per destination node; each thread owns 2 features
// (float2), walks the node's CSR in-edge list accumulating in registers,
// then writes the post-activation row once into the JK-concat buffer.
// ---------------------------------------------------------------------------
__global__ __launch_bounds__(256) void
agg_bn_relu_kernel(const float* __restrict__ m,
                   const int* __restrict__ row_ptr,
                   const int* __restrict__ col,
                   const float* __restrict__ norm_dst,
                   const float* __restrict__ b,
                   const float* __restrict__ gamma,
                   const float* __restrict__ beta,
                   const float* __restrict__ mean,
                   const float* __restrict__ var,
                   float* __restrict__ hout /* hs + i*H, row stride LH */) {
  const int n = blockIdx.x;
  const int f = (int)threadIdx.x * 2;
  const int e0 = row_ptr[n];
  const int e1 = row_ptr[n + 1];

  float ax = 0.0f, ay = 0.0f;
  for (int j = e0; j < e1; ++j) {
    const int s = col[j];
    const float2 v = *(const float2*)(m + (size_t)s * HH + f);
    ax += v.x;
    ay += v.y;
  }

  const float  nd = norm_dst[n];
  const float2 bb = *(const float2*)(b + f);
  const float2 gm = *(const float2*)(gamma + f);
  const float2 bt = *(const float2*)(beta + f);
  const float2 mu = *(const float2*)(mean + f);
  const float2 va = *(const float2*)(var + f);

  float vx = (ax * nd + bb.x - mu.x) * (gm.x * rsqrtf(va.x + EPSB)) + bt.x;
  float vy = (ay * nd + bb.y - mu.y) * (gm.y * rsqrtf(va.y + EPSB)) + bt.y;

  float* __restrict__ hp = hout + (size_t)n * LH + f;
  hp[0] = vx > 0.0f ? vx : 0.0f;
  hp[1] = vy > 0.0f ? vy : 0.0f;
}

// ---------------------------------------------------------------------------
// Host-side launch
// ---------------------------------------------------------------------------
extern "C" void kernel_launch(void* const* d_in, const int* in_sizes, int n_in,
                              void* d_out, int out_size, void* d_ws, size_t ws_size,
                              hipStream_t stream) {
  const float* x        = (const float*)d_in[0];   // [N, IN]
  const float* W0       = (const float*)d_in[1];   // [IN, H]
  const float* Ws       = (const float*)d_in[2];   // [L-1, H, H]
  const float* bs       = (const float*)d_in[3];   // [L, H]
  const float* gamma    = (const float*)d_in[4];   // [L, H]
  const float* beta     = (const float*)d_in[5];   // [L, H]
  const float* run_mean = (const float*)d_in[6];   // [L, H]
  const float* run_var  = (const float*)d_in[7];   // [L, H]
  const float* linW     = (const float*)d_in[8];   // [L*H, C]
  const float* linb     = (const float*)d_in[9];   // [C]
  const int*   src      = (const int*)d_in[10];    // [E]
  const int*   dst      = (const int*)d_in[11];    // [E]
  float*       out      = (float*)d_out;           // [N, C]

  // Workspace layout:
  //   ocnt[N] icnt[N] row_ptr[N+1] cursor[N] col[E]      (int)
  //   norm_src[N] norm_dst[N] wT[512*512] m[N*H] hs[N*L*H] (float)
  int*   ocnt     = (int*)d_ws;
  int*   icnt     = ocnt + NN;
  int*   row_ptr  = icnt + NN;
  int*   cursor   = row_ptr + NN + 1;
  int*   col      = cursor + NN;
  float* norm_src = (float*)(col + EE);
  float* norm_dst = norm_src + NN;
  float* wT_buf   = norm_dst + NN;                 // reused per GEMM (<= 1 MB)
  float* m_buf    = wT_buf + (size_t)HH * HH;
  float* hs_buf   = m_buf + (size_t)NN * HH;       // [N, L*H] JK-concat

  // --- degrees, norms, CSR build (once per call) ---
  hipMemsetAsync(ocnt, 0, 2 * NN * sizeof(int), stream);
  count_kernel<<<(EE + 255) / 256, 256, 0, stream>>>(src, dst, ocnt, icnt, EE);
  norm_from_cnt<<<(2 * NN + 255) / 256, 256, 0, stream>>>(ocnt, norm_src, 2 * NN);
  scan_kernel<<<1, 256, 0, stream>>>(icnt, row_ptr, cursor, NN);
  fill_kernel<<<(EE + 255) / 256, 256, 0, stream>>>(src, dst, cursor, col, EE);

  // --- 6 GraphConv layers ---
  for (int i = 0; i < LL; ++i) {
    const float* Ain = (i == 0) ? x : (hs_buf + (size_t)(i - 1) * HH);
    const int    lda = (i == 0) ? INF : LH;
    const float* W   = (i == 0) ? W0 : (Ws + (size_t)(i - 1) * HH * HH);
    const int    K   = (i == 0) ? INF : HH;

    // WT = W^T  (K x H -> H x K)
    transpose_kernel<<<dim3(HH / 32, K / 32), 256, 0, stream>>>(W, K, HH, wT_buf);

    // m = diag(norm_src) * Ain @ W
    wmma_gemm_f32<<<dim3(NN / MB, HH / NB), MB / 16 * 32, 0, stream>>>(
        Ain, lda, norm_src, wT_buf, HH, m_buf, HH, nullptr, K);

    // h = relu(BN(segment_sum(m[src],dst) * norm_dst + b)) -> hs[:, i*H:(i+1)*H]
    agg_bn_relu_kernel<<<NN, 256, 0, stream>>>(
        m_buf, row_ptr, col, norm_dst, bs + i * HH, gamma + i * HH,
        beta + i * HH, run_mean + i * HH, run_var + i * HH,
        hs_buf + (size_t)i * HH);
  }

  // --- JK head: out = hs @ linW + linb ---
  transpose_kernel<<<dim3(CC / 32, LH / 32), 256, 0, stream>>>(linW, LH, CC, wT_buf);
  wmma_gemm_f32<<<dim3(NN / MB, CC / NB), MB / 16 * 32, 0, stream>>>(
      hs_buf, LH, nullptr, wT_buf, CC, out, CC, linb, LH);
}